// SimpleModel_Onset_41566693491378
// MI455X (gfx1250) — compile-verified
//
#include <hip/hip_runtime.h>
#include <hip/hip_bf16.h>
#include <math.h>

// ---------------- problem constants ----------------
#define N_BINS 229
#define KPAD   232            // N_BINS padded to multiple of 4
#define OUT_F  88
#define MODEL  128
#define W_SIZE 30
#define WIN    61             // 2*W_SIZE+1
#define BATCH  4
#define TLEN   1024
#define ROWS   (BATCH * TLEN) // 4096, multiple of 16

typedef float v2f __attribute__((ext_vector_type(2)));
typedef float v8f __attribute__((ext_vector_type(8)));

__device__ __forceinline__ float fast_sigmoid(float x) {
    // v_exp_f32 + v_rcp_f32 (1-ulp HW reciprocal) instead of IEEE div chain
    return __builtin_amdgcn_rcpf(1.0f + __expf(-x));
}

// ---------------------------------------------------------------------------
// fp32 WMMA GEMM:  C[M,N] = act( A[M,Kpad] * B[Kpad,Npad] + Cin + bias[N] )
// One wave per 16x16 tile, V_WMMA_F32_16X16X4_F32 per 4-wide K step.
// A: padded lda (even), zero-filled K tail  -> unconditional b64 loads.
// Bp: pre-packed pair-interleaved layout Bp[k/2][n][k%2], zero padded
//     -> unconditional b64 loads, no guards anywhere in the hot loop.
// act: 0 = identity, 1 = sigmoid
// ---------------------------------------------------------------------------
__global__ __launch_bounds__(32)
void gemm_wmma_f32(const float* __restrict__ A, int lda,
                   const float* __restrict__ Bp, int Npad,
                   float* __restrict__ C, int ldc,
                   const float* __restrict__ Cin, int ldcin,
                   const float* __restrict__ bias,
                   int N, int Kpad, int act)
{
    const int lane  = threadIdx.x & 31;
    const int mtile = blockIdx.y;
    const int ntile = blockIdx.x;
    const int half  = lane >> 4;                 // 0: K pair 0, 1: K pair 1
    const int mrow  = mtile * 16 + (lane & 15);
    const int ncol  = ntile * 16 + (lane & 15);

    // lane's streaming pointers (both 8-byte aligned by construction)
    const float* ap = A  + (size_t)mrow * lda + 2 * half;
    const float* bp = Bp + ((size_t)half * Npad + ncol) * 2;

    v8f acc = {};
    if (Cin) {
        const bool nok = (ncol < N);
        #pragma unroll
        for (int j = 0; j < 8; ++j) {
            int row = mtile * 16 + j + half * 8;
            acc[j] = nok ? Cin[(size_t)row * ldcin + ncol] : 0.0f;
        }
    }

    const int ksteps = Kpad >> 2;
    #pragma unroll 2
    for (int s = 0; s < ksteps; ++s) {
        v2f a = *reinterpret_cast<const v2f*>(ap);
        v2f b = *reinterpret_cast<const v2f*>(bp);
        acc = __builtin_amdgcn_wmma_f32_16x16x4_f32(
                  false, a, false, b, (short)0, acc, false, false);
        ap += 4;
        bp += (size_t)4 * Npad;
    }

    if (ncol < N) {
        const float bv = bias ? bias[ncol] : 0.0f;
        #pragma unroll
        for (int j = 0; j < 8; ++j) {
            int row = mtile * 16 + j + half * 8;
            float x = acc[j] + bv;
            if (act == 1) x = fast_sigmoid(x);
            C[(size_t)row * ldc + ncol] = x;
        }
    }
}

// ---------------------------------------------------------------------------
// Pack weight B[K,N] (row major, ld=N) into padded pair-interleaved
// Bp[Kpad/2][Npad][2]; zero fill K/N tails.  Output linear index == idx.
// ---------------------------------------------------------------------------
__global__ void pack_b_kernel(const float* __restrict__ B, int K, int N,
                              float* __restrict__ Bp, int Kpad, int Npad)
{
    int idx = blockIdx.x * blockDim.x + threadIdx.x;
    int total = Kpad * Npad;
    if (idx >= total) return;
    int j  = idx & 1;
    int n  = (idx >> 1) % Npad;
    int kk = idx / (2 * Npad);
    int k  = 2 * kk + j;
    Bp[idx] = (k < K && n < N) ? B[(size_t)k * N + n] : 0.0f;
}

// spec [ROWS,229] -> spec_pad [ROWS,232] with zero tail
__global__ void pad_spec_kernel(const float* __restrict__ spec,
                                float* __restrict__ sp, int total)
{
    int idx = blockIdx.x * blockDim.x + threadIdx.x;
    if (idx >= total) return;
    int r = idx / KPAD, c = idx % KPAD;
    sp[idx] = (c < N_BINS) ? spec[(size_t)r * N_BINS + c] : 0.0f;
}

// ---------------------------------------------------------------------------
// Factored windowed attention + weighted sum.
//   score[w] = v . tanh( Q[r] + ba + P[b, t+w-30] )   (P term = 0 if padded)
//   a = softmax(score);  wsum[r,f] = sum_w a[w] * spec[b, t+w-30, f]
// wsum written with padded leading dim KPAD (zero tail) for the next GEMM.
// ---------------------------------------------------------------------------
__global__ __launch_bounds__(64)
void attend_kernel(const float* __restrict__ Q,    // [ROWS, MODEL]
                   const float* __restrict__ P,    // [ROWS, MODEL]
                   const float* __restrict__ ba,   // [MODEL]
                   const float* __restrict__ v,    // [MODEL]
                   const float* __restrict__ spec, // [ROWS, N_BINS]
                   float* __restrict__ a_out,      // [ROWS, WIN]
                   float* __restrict__ wsum,       // [ROWS, KPAD]
                   int T)
{
    __shared__ float sQ[MODEL];
    __shared__ float sV[MODEL];
    __shared__ float sS[WIN];
    __shared__ float sMax, sInv;

    const int r   = blockIdx.x;
    const int b   = r / T;
    const int t   = r % T;
    const int tid = threadIdx.x;

    for (int i = tid; i < MODEL; i += 64) {
        sQ[i] = Q[(size_t)r * MODEL + i] + ba[i];
        sV[i] = v[i];
    }
    __syncthreads();

    if (tid < WIN) {
        const int t2 = t + tid - W_SIZE;
        float s = 0.0f;
        if (t2 >= 0 && t2 < T) {
            const float* Pr = P + (size_t)(b * T + t2) * MODEL;
            for (int i = 0; i < MODEL; ++i) s += sV[i] * tanhf(sQ[i] + Pr[i]);
        } else {
            for (int i = 0; i < MODEL; ++i) s += sV[i] * tanhf(sQ[i]);
        }
        sS[tid] = s;
    }
    __syncthreads();

    if (tid == 0) {
        float mx = sS[0];
        for (int w = 1; w < WIN; ++w) mx = fmaxf(mx, sS[w]);
        float sm = 0.0f;
        for (int w = 0; w < WIN; ++w) sm += __expf(sS[w] - mx);
        sMax = mx;
        sInv = __builtin_amdgcn_rcpf(sm);
    }
    __syncthreads();

    if (tid < WIN) {
        float aw = __expf(sS[tid] - sMax) * sInv;
        sS[tid] = aw;
        a_out[(size_t)r * WIN + tid] = aw;
    }
    __syncthreads();

    for (int f = tid; f < KPAD; f += 64) {
        float acc = 0.0f;
        if (f < N_BINS) {
            for (int w = 0; w < WIN; ++w) {
                const int t2 = t + w - W_SIZE;
                if (t2 >= 0 && t2 < T)
                    acc += sS[w] * spec[(size_t)(b * T + t2) * N_BINS + f];
            }
        }
        wsum[(size_t)r * KPAD + f] = acc;
    }
}

// x[r, 0:88] = feat_pred[r]; x[r, 88:176] = onset_pred[r]
__global__ void build_x_kernel(const float* __restrict__ feat,
                               const float* __restrict__ onset,
                               float* __restrict__ x, int total)
{
    int idx = blockIdx.x * blockDim.x + threadIdx.x;
    if (idx < total) {
        int r = idx / (2 * OUT_F);
        int c = idx % (2 * OUT_F);
        x[idx] = (c < OUT_F) ? feat[(size_t)r * OUT_F + c]
                             : onset[(size_t)r * OUT_F + (c - OUT_F)];
    }
}

// ---------------------------------------------------------------------------
extern "C" void kernel_launch(void* const* d_in, const int* in_sizes, int n_in,
                              void* d_out, int out_size, void* d_ws, size_t ws_size,
                              hipStream_t stream)
{
    (void)in_sizes; (void)n_in; (void)out_size; (void)ws_size;

    const float* spec    = (const float*)d_in[0];   // [4,1024,229]
    const float* W_frame = (const float*)d_in[1];   // [229,88]
    const float* b_frame = (const float*)d_in[2];   // [88]
    const float* W_atno  = (const float*)d_in[3];   // [458,128]
    const float* b_atno  = (const float*)d_in[4];   // [128]
    const float* v_o     = (const float*)d_in[5];   // [128]
    const float* W_lin1  = (const float*)d_in[6];   // [229,88]
    const float* b_lin1  = (const float*)d_in[7];   // [88]
    const float* W_atnc  = (const float*)d_in[8];   // [405,128]
    const float* b_atnc  = (const float*)d_in[9];   // [128]
    const float* v_c     = (const float*)d_in[10];  // [128]
    const float* W_linc  = (const float*)d_in[11];  // [405,88]
    const float* b_linc  = (const float*)d_in[12];  // [88]

    // ---- output regions (flat concat in return order) ----
    float* out         = (float*)d_out;
    const size_t n_fp  = (size_t)ROWS * OUT_F;
    const size_t n_att = (size_t)ROWS * WIN;
    float* o_frame  = out;                          // frame_pred
    float* o_aframe = out + n_fp;                   // a_frame
    float* o_onset  = out + n_fp + n_att;           // onset_pred
    float* o_aonset = out + 2 * n_fp + n_att;       // a_onset
    float* o_feat   = out + 2 * n_fp + 2 * n_att;   // feat_pred

    // ---- workspace layout (floats; all block sizes even -> 8B alignment) ----
    float* ws = (float*)d_ws;
    size_t off = 0;
    float* sp  = ws + off; off += (size_t)ROWS * KPAD;     // padded spec
    float* Qo  = ws + off; off += (size_t)ROWS * MODEL;
    float* Po  = ws + off; off += (size_t)ROWS * MODEL;
    float* Pc  = ws + off; off += (size_t)ROWS * MODEL;
    float* wo  = ws + off; off += (size_t)ROWS * KPAD;     // weighted_o (padded)
    float* xb  = ws + off; off += (size_t)ROWS * 2 * OUT_F;
    float* Qc  = ws + off; off += (size_t)ROWS * MODEL;
    float* Fx  = ws + off; off += (size_t)ROWS * OUT_F;    // x @ W_linc_top
    float* wc  = ws + off; off += (size_t)ROWS * KPAD;     // weighted_c (padded)
    float* BpFrame = ws + off; off += (size_t)KPAD * 96;
    float* BpQo    = ws + off; off += (size_t)KPAD * 128;
    float* BpPo    = ws + off; off += (size_t)KPAD * 128;
    float* BpPc    = ws + off; off += (size_t)KPAD * 128;
    float* BpLin1  = ws + off; off += (size_t)KPAD * 96;
    float* BpQc    = ws + off; off += (size_t)176  * 128;
    float* BpFxW   = ws + off; off += (size_t)176  * 96;
    float* BpLcBot = ws + off; off += (size_t)KPAD * 96;

    // ---- stage 0: pad A operand, pack all B operands ----
    {
        int tot = ROWS * KPAD;
        pad_spec_kernel<<<(tot + 255) / 256, 256, 0, stream>>>(spec, sp, tot);
    }
    auto packs = [&](const float* B, int K, int N, float* Bp, int Kp, int Np) {
        int tot = Kp * Np;
        pack_b_kernel<<<(tot + 255) / 256, 256, 0, stream>>>(B, K, N, Bp, Kp, Np);
    };
    packs(W_frame,                              N_BINS, OUT_F, BpFrame, KPAD, 96);
    packs(W_atno,                               N_BINS, MODEL, BpQo,    KPAD, 128);
    packs(W_atno + (size_t)N_BINS * MODEL,      N_BINS, MODEL, BpPo,    KPAD, 128);
    packs(W_atnc + (size_t)(2 * OUT_F) * MODEL, N_BINS, MODEL, BpPc,    KPAD, 128);
    packs(W_lin1,                               N_BINS, OUT_F, BpLin1,  KPAD, 96);
    packs(W_atnc,                               2 * OUT_F, MODEL, BpQc, 176,  128);
    packs(W_linc,                               2 * OUT_F, OUT_F, BpFxW, 176, 96);
    packs(W_linc + (size_t)(2 * OUT_F) * OUT_F, N_BINS, OUT_F, BpLcBot, KPAD, 96);

    const dim3 blk(32);
    const int MT = ROWS / 16;              // 256 M-tiles
    const dim3 g88(6, MT);                 // Npad=96  -> 6 tiles
    const dim3 g128(8, MT);                // Npad=128 -> 8 tiles

    // 1) feat_pred = spec @ W_frame + b_frame
    gemm_wmma_f32<<<g88, blk, 0, stream>>>(sp, KPAD, BpFrame, 96,
        o_feat, OUT_F, nullptr, 0, b_frame, OUT_F, KPAD, 0);
    // 2) Q_o = spec @ W_attn_o[0:229]
    gemm_wmma_f32<<<g128, blk, 0, stream>>>(sp, KPAD, BpQo, 128,
        Qo, MODEL, nullptr, 0, nullptr, MODEL, KPAD, 0);
    // 3) P_o = spec @ W_attn_o[229:458]
    gemm_wmma_f32<<<g128, blk, 0, stream>>>(sp, KPAD, BpPo, 128,
        Po, MODEL, nullptr, 0, nullptr, MODEL, KPAD, 0);
    // 4) P_c = spec @ W_attn_c[176:405]
    gemm_wmma_f32<<<g128, blk, 0, stream>>>(sp, KPAD, BpPc, 128,
        Pc, MODEL, nullptr, 0, nullptr, MODEL, KPAD, 0);

    // 5) onset attention -> a_onset, weighted_o
    attend_kernel<<<ROWS, 64, 0, stream>>>(Qo, Po, b_atno, v_o, spec,
                                           o_aonset, wo, TLEN);

    // 6) onset_pred = sigmoid(weighted_o @ W_lin1 + b_lin1)
    gemm_wmma_f32<<<g88, blk, 0, stream>>>(wo, KPAD, BpLin1, 96,
        o_onset, OUT_F, nullptr, 0, b_lin1, OUT_F, KPAD, 1);

    // 7) x = [feat_pred | onset_pred]
    {
        int total = ROWS * 2 * OUT_F;
        build_x_kernel<<<(total + 255) / 256, 256, 0, stream>>>(o_feat, o_onset, xb, total);
    }

    // 8) Q_c = x @ W_attn_c[0:176]
    gemm_wmma_f32<<<g128, blk, 0, stream>>>(xb, 2 * OUT_F, BpQc, 128,
        Qc, MODEL, nullptr, 0, nullptr, MODEL, 176, 0);
    // 9) Fx = x @ W_linc[0:176]
    gemm_wmma_f32<<<g88, blk, 0, stream>>>(xb, 2 * OUT_F, BpFxW, 96,
        Fx, OUT_F, nullptr, 0, nullptr, OUT_F, 176, 0);

    // 10) combine attention -> a_frame, weighted_c
    attend_kernel<<<ROWS, 64, 0, stream>>>(Qc, Pc, b_atnc, v_c, spec,
                                           o_aframe, wc, TLEN);

    // 11) frame_pred = sigmoid(Fx + weighted_c @ W_linc[176:405] + b_linc)
    gemm_wmma_f32<<<g88, blk, 0, stream>>>(wc, KPAD, BpLcBot, 96,
        o_frame, OUT_F, Fx, OUT_F, b_linc, OUT_F, KPAD, 1);
}